// _Rnet_post_27178553049679
// MI455X (gfx1250) — compile-verified
//
#include <hip/hip_runtime.h>

// _Rnet_post: element-wise bbox regression + threshold mask.
// Memory-bound streaming kernel: 64 B/box, 268 MB total -> ~11.5 us at 23.3 TB/s.
// No matrix math in this op => no WMMA by roofline arithmetic; optimize HBM path:
// b64/b128 NT loads, b128 NT stores, WGP-scope global_prefetch_b8 one iter ahead,
// uniform fast path for full blocks (no per-iteration exec-mask churn).

typedef float v2f __attribute__((ext_vector_type(2)));
typedef float v4f __attribute__((ext_vector_type(4)));

#define BLOCK 256
#define ITER 4

__device__ __forceinline__ void process_one(
    long long i,
    const float* __restrict__ cls, const float* __restrict__ reg,
    const float* __restrict__ rct, float fh, float fw,
    float* __restrict__ out_rects, float* __restrict__ out_scores,
    float* __restrict__ out_keep)
{
    // Streaming (non-temporal) vector loads: b64 + b128 + b128 per box.
    const v2f c = __builtin_nontemporal_load((const v2f*)(cls + 2 * i));
    const v4f d = __builtin_nontemporal_load((const v4f*)(reg + 4 * i));
    const v4f r = __builtin_nontemporal_load((const v4f*)(rct + 4 * i));

    const float score = c.y;                        // P(face)
    const float keep  = (score > 0.7f) ? 1.0f : 0.0f;

    const float w = r.z - r.x;
    const float h = r.w - r.y;

    // regression + clip to image bounds (v_fmac + v_max_num/v_min_num)
    const float nx1 = fminf(fmaxf(fmaf(d.x, w, r.x), 0.0f), fw);
    const float ny1 = fminf(fmaxf(fmaf(d.y, h, r.y), 0.0f), fh);
    const float nx2 = fminf(fmaxf(fmaf(d.z, w, r.z), 0.0f), fw);
    const float ny2 = fminf(fmaxf(fmaf(d.w, h, r.w), 0.0f), fh);

    v4f o;
    o.x = nx1 * keep;
    o.y = ny1 * keep;
    o.z = nx2 * keep;
    o.w = ny2 * keep;

    __builtin_nontemporal_store(o, (v4f*)(out_rects + 4 * i));
    __builtin_nontemporal_store(score * keep, out_scores + i);
    __builtin_nontemporal_store(keep, out_keep + i);
}

__global__ __launch_bounds__(BLOCK) void rnet_post_kernel(
    const float* __restrict__ cls,      // [n,2] classifier
    const float* __restrict__ reg,      // [n,4] bbox_regress
    const float* __restrict__ rct,      // [n,4] input_rects (x1,y1,x2,y2)
    const int*   __restrict__ pH,       // scalar input_height
    const int*   __restrict__ pW,       // scalar input_width
    float* __restrict__ out_rects,      // [n,4]
    float* __restrict__ out_scores,     // [n]
    float* __restrict__ out_keep,       // [n]
    long long n)
{
    const float fh = (float)(*pH);
    const float fw = (float)(*pW);

    // Each block owns a contiguous chunk of BLOCK*ITER boxes; thread t does
    // boxes chunk + t + k*BLOCK (coalesced within each iteration).
    const long long chunk = (long long)blockIdx.x * (BLOCK * ITER);
    const long long base  = chunk + threadIdx.x;

    if (chunk + (long long)(BLOCK * ITER) <= n) {
        // ---- Fast path: whole block in bounds, straight-line unrolled ----
#pragma unroll
        for (int k = 0; k < ITER; ++k) {
            const long long i = base + (long long)k * BLOCK;

            // WGP-scope prefetch of next iteration (pull into all cache levels).
            if (k + 1 < ITER) {
                const long long j = i + BLOCK;
                __builtin_prefetch(cls + 2 * j, 0, 3);
                __builtin_prefetch(reg + 4 * j, 0, 3);
                __builtin_prefetch(rct + 4 * j, 0, 3);
            }

            process_one(i, cls, reg, rct, fh, fw,
                        out_rects, out_scores, out_keep);
        }
    } else {
        // ---- Tail path: per-element guard ----
#pragma unroll
        for (int k = 0; k < ITER; ++k) {
            const long long i = base + (long long)k * BLOCK;
            if (i < n) {
                process_one(i, cls, reg, rct, fh, fw,
                            out_rects, out_scores, out_keep);
            }
        }
    }
}

extern "C" void kernel_launch(void* const* d_in, const int* in_sizes, int n_in,
                              void* d_out, int out_size, void* d_ws, size_t ws_size,
                              hipStream_t stream) {
    (void)n_in; (void)out_size; (void)d_ws; (void)ws_size;

    const float* cls = (const float*)d_in[0];   // [B,N,2]
    const float* reg = (const float*)d_in[1];   // [B,N,4]
    const float* rct = (const float*)d_in[2];   // [B,N,4]
    const int*   pH  = (const int*)d_in[3];     // scalar
    const int*   pW  = (const int*)d_in[4];     // scalar

    const long long n = (long long)in_sizes[0] / 2;  // B*N boxes

    // d_out layout: rects [n,4] | scores [n] | keep [n] (reference return order)
    float* out_rects  = (float*)d_out;
    float* out_scores = out_rects + 4 * n;
    float* out_keep   = out_scores + n;

    const long long per_block = (long long)BLOCK * ITER;
    const int grid = (int)((n + per_block - 1) / per_block);

    rnet_post_kernel<<<dim3(grid), dim3(BLOCK), 0, stream>>>(
        cls, reg, rct, pH, pW, out_rects, out_scores, out_keep, n);
}